// IIRFilter_12575664243046
// MI455X (gfx1250) — compile-verified
//
#include <hip/hip_runtime.h>
#include <cmath>

typedef __attribute__((ext_vector_type(2))) float v2f;
typedef __attribute__((ext_vector_type(8))) float v8f;

#define NSAMP           131072
#define NBATCH          64
#define BLK_SPAN        8192
#define NBLK_PER_BATCH  16
#define NBLOCKS         (NBATCH * NBLK_PER_BATCH)   // 1024
#define T_SPAN          32                          // samples per thread (phase A)
#define WAVE_SPAN       1024                        // samples per wave (phase B)
#define GROUPS_PER_WAVE (WAVE_SPAN / 256)           // 4 groups of 16 chunks x 16 samples

struct FConsts {
    float A[4];          // per-sample transition 2H - I (row major)
    float b2[2];         // per-sample input gain 2*gHB
    float P[8][4];       // A^(32*2^l), l=0..7   (256-thread LDS scan)
    float A16o[4][4];    // A^(16*o), o=1,2,4,8  (16-lane chunk scan / A^16j build)
    float A256[4];       // group advance
    float A1024p[8][4];  // A^(1024*w), w=0..7   (wave start within block)
    float A8192[4];      // block advance
    float W0[16], W1[16];// rows of W: c_chunk = sum_k A^(15-k) b x[k]
    float L[16][16];     // intra-chunk impulse matrix (diag = d)
    float G0[16], G1[16];// G[i] = h^T A^i
};

// ---------------- kernel 1: per-block affine constants + per-wave prefixes ----------------
__global__ __launch_bounds__(256)
void k_blockconst(const float* __restrict__ x, float* __restrict__ cblk,
                  float* __restrict__ ewave, FConsts C) {
    const int bid = blockIdx.x;
    const int t   = threadIdx.x;
    const size_t base = (size_t)bid * BLK_SPAN;

    float s0 = 0.f, s1 = 0.f;
    const float4* xp = reinterpret_cast<const float4*>(x + base + (size_t)t * T_SPAN);
#pragma unroll
    for (int i = 0; i < T_SPAN / 4; ++i) {
        float4 v = xp[i];
        { float n0=fmaf(C.A[0],s0,fmaf(C.A[1],s1,C.b2[0]*v.x)); float n1=fmaf(C.A[2],s0,fmaf(C.A[3],s1,C.b2[1]*v.x)); s0=n0; s1=n1; }
        { float n0=fmaf(C.A[0],s0,fmaf(C.A[1],s1,C.b2[0]*v.y)); float n1=fmaf(C.A[2],s0,fmaf(C.A[3],s1,C.b2[1]*v.y)); s0=n0; s1=n1; }
        { float n0=fmaf(C.A[0],s0,fmaf(C.A[1],s1,C.b2[0]*v.z)); float n1=fmaf(C.A[2],s0,fmaf(C.A[3],s1,C.b2[1]*v.z)); s0=n0; s1=n1; }
        { float n0=fmaf(C.A[0],s0,fmaf(C.A[1],s1,C.b2[0]*v.w)); float n1=fmaf(C.A[2],s0,fmaf(C.A[3],s1,C.b2[1]*v.w)); s0=n0; s1=n1; }
    }

    __shared__ float sh0[256], sh1[256];
    sh0[t] = s0; sh1[t] = s1;
    __syncthreads();
#pragma unroll
    for (int l = 0; l < 8; ++l) {
        int off = 1 << l;
        float p0 = 0.f, p1 = 0.f;
        bool ok = (t >= off);
        if (ok) { p0 = sh0[t - off]; p1 = sh1[t - off]; }
        __syncthreads();
        if (ok) {
            s0 = fmaf(C.P[l][0], p0, fmaf(C.P[l][1], p1, s0));
            s1 = fmaf(C.P[l][2], p0, fmaf(C.P[l][3], p1, s1));
            sh0[t] = s0; sh1[t] = s1;
        }
        __syncthreads();
    }

    // per-wave exclusive prefixes for kernel 3 (inclusive of thread 32w-1)
    if (t == 0) {
        ewave[(size_t)bid * 16 + 0] = 0.f;
        ewave[(size_t)bid * 16 + 1] = 0.f;
    }
    if ((t & 31) == 31 && t != 255) {
        int w = (t >> 5) + 1;          // 1..7
        ewave[(size_t)bid * 16 + 2 * w]     = s0;
        ewave[(size_t)bid * 16 + 2 * w + 1] = s1;
    }
    if (t == 255) { cblk[2 * bid] = s0; cblk[2 * bid + 1] = s1; }
}

// ---------------- kernel 2: scan block constants per batch ----------------
__global__ void k_scanblocks(const float* __restrict__ cblk, float* __restrict__ sblk, FConsts C) {
    int b = threadIdx.x;
    if (b >= NBATCH) return;
    float s0 = 0.f, s1 = 0.f;
    int base = b * NBLK_PER_BATCH;
    for (int k = 0; k < NBLK_PER_BATCH; ++k) {
        sblk[2 * (base + k)]     = s0;
        sblk[2 * (base + k) + 1] = s1;
        float c0 = cblk[2 * (base + k)], c1 = cblk[2 * (base + k) + 1];
        float n0 = fmaf(C.A8192[0], s0, fmaf(C.A8192[1], s1, c0));
        float n1 = fmaf(C.A8192[2], s0, fmaf(C.A8192[3], s1, c1));
        s0 = n0; s1 = n1;
    }
}

// ---------------- kernel 3: WMMA output pass (single read of x) ----------------
__global__ __launch_bounds__(256)
void k_output(const float* __restrict__ x, const float* __restrict__ sblk,
              const float* __restrict__ ewave, float* __restrict__ out, FConsts C) {
    const int bid = blockIdx.x;
    const int t   = threadIdx.x;
    const size_t base = (size_t)bid * BLK_SPAN;

    const int w    = t >> 5;
    const int lane = t & 31;
    const bool lo  = lane < 16;
    const int  j   = lane & 15;      // chunk column (B/D N index)

    // ---- wave starting state: A^(1024w) * s_block + ewave[bid][w] ----
    float sb0 = sblk[2 * bid], sb1 = sblk[2 * bid + 1];
    float sw0 = fmaf(C.A1024p[w][0], sb0, C.A1024p[w][1] * sb1) + ewave[(size_t)bid * 16 + 2 * w];
    float sw1 = fmaf(C.A1024p[w][2], sb0, C.A1024p[w][3] * sb1) + ewave[(size_t)bid * 16 + 2 * w + 1];

    // ---- constant A-operands: L split into four 16x4 K-blocks ----
    v2f am[4];
#pragma unroll
    for (int m = 0; m < 4; ++m) {
        int cb = 4 * m + (lo ? 0 : 2);
        am[m].x = C.L[j][cb];
        am[m].y = C.L[j][cb + 1];
    }
    // W slots aligned with this lane's K-set
    float w0s[8], w1s[8];
#pragma unroll
    for (int m = 0; m < 4; ++m) {
        w0s[2*m]   = lo ? C.W0[4*m]   : C.W0[4*m+2];
        w0s[2*m+1] = lo ? C.W0[4*m+1] : C.W0[4*m+3];
        w1s[2*m]   = lo ? C.W1[4*m]   : C.W1[4*m+2];
        w1s[2*m+1] = lo ? C.W1[4*m+1] : C.W1[4*m+3];
    }
    // G rows handled by this lane's D registers (M = r or r+8)
    float g0s[8], g1s[8];
#pragma unroll
    for (int r = 0; r < 8; ++r) {
        g0s[r] = lo ? C.G0[r] : C.G0[r + 8];
        g1s[r] = lo ? C.G1[r] : C.G1[r + 8];
    }

    const float* xw = x   + base + (size_t)w * WAVE_SPAN;
    float*       ow = out + base + (size_t)w * WAVE_SPAN;

    for (int grp = 0; grp < GROUPS_PER_WAVE; ++grp) {
        const float* xg = xw + grp * 256;

        // load X (16 samples x 16 chunks) straight into B-operand layout
        v2f bm[4];
#pragma unroll
        for (int m = 0; m < 4; ++m) {
            int kb = 4 * m + (lo ? 0 : 2);
            float2 vv = *reinterpret_cast<const float2*>(xg + j * 16 + kb);
            bm[m].x = vv.x; bm[m].y = vv.y;
        }

        // O = L * X  via 4 chained f32 WMMAs (K = 16)
        v8f acc = {0.f,0.f,0.f,0.f,0.f,0.f,0.f,0.f};
        acc = __builtin_amdgcn_wmma_f32_16x16x4_f32(false, am[0], false, bm[0], (short)0, acc, false, false);
        acc = __builtin_amdgcn_wmma_f32_16x16x4_f32(false, am[1], false, bm[1], (short)0, acc, false, false);
        acc = __builtin_amdgcn_wmma_f32_16x16x4_f32(false, am[2], false, bm[2], (short)0, acc, false, false);
        acc = __builtin_amdgcn_wmma_f32_16x16x4_f32(false, am[3], false, bm[3], (short)0, acc, false, false);

        // per-chunk constants c_j = W x_j (half per lane group, combine across halves)
        float p0 = 0.f, p1 = 0.f;
#pragma unroll
        for (int m = 0; m < 4; ++m) {
            p0 = fmaf(w0s[2*m], bm[m].x, fmaf(w0s[2*m+1], bm[m].y, p0));
            p1 = fmaf(w1s[2*m], bm[m].x, fmaf(w1s[2*m+1], bm[m].y, p1));
        }
        float c0 = p0 + __shfl_xor(p0, 16, 32);
        float c1 = p1 + __shfl_xor(p1, 16, 32);

        // inclusive affine scan over 16 chunk columns
        float q0 = c0, q1 = c1;
#pragma unroll
        for (int lvl = 0; lvl < 4; ++lvl) {
            int off = 1 << lvl;
            float t0 = __shfl_up(q0, off, 16);
            float t1 = __shfl_up(q1, off, 16);
            bool ok = (j >= off);
            float n0 = fmaf(C.A16o[lvl][0], t0, fmaf(C.A16o[lvl][1], t1, q0));
            float n1 = fmaf(C.A16o[lvl][2], t0, fmaf(C.A16o[lvl][3], t1, q1));
            q0 = ok ? n0 : q0;
            q1 = ok ? n1 : q1;
        }
        // exclusive prefix -> chunk-start state contribution
        float e0 = __shfl_up(q0, 1, 16);
        float e1 = __shfl_up(q1, 1, 16);
        e0 = (j == 0) ? 0.f : e0;
        e1 = (j == 0) ? 0.f : e1;

        // A^(16j) * s_wave by binary decomposition of j
        float v0 = sw0, v1 = sw1;
#pragma unroll
        for (int lvl = 0; lvl < 4; ++lvl) {
            bool bit = ((j >> lvl) & 1) != 0;
            float n0 = fmaf(C.A16o[lvl][0], v0, C.A16o[lvl][1] * v1);
            float n1 = fmaf(C.A16o[lvl][2], v0, C.A16o[lvl][3] * v1);
            v0 = bit ? n0 : v0;
            v1 = bit ? n1 : v1;
        }
        float st0 = v0 + e0, st1 = v1 + e1;

        // add state contribution G * s_j to the WMMA result
#pragma unroll
        for (int r = 0; r < 8; ++r)
            acc[r] = fmaf(g0s[r], st0, fmaf(g1s[r], st1, acc[r]));

        // store D: rows 0-7 (lanes 0-15) / rows 8-15 (lanes 16-31), contiguous
        float* og = ow + grp * 256 + j * 16 + (lo ? 0 : 8);
        *reinterpret_cast<float4*>(og)     = make_float4(acc[0], acc[1], acc[2], acc[3]);
        *reinterpret_cast<float4*>(og + 4) = make_float4(acc[4], acc[5], acc[6], acc[7]);

        // advance wave state by one group: s = A^256 s + q_15
        float r0 = __shfl(q0, 15, 16);
        float r1 = __shfl(q1, 15, 16);
        float ns0 = fmaf(C.A256[0], sw0, fmaf(C.A256[1], sw1, r0));
        float ns1 = fmaf(C.A256[2], sw0, fmaf(C.A256[3], sw1, r1));
        sw0 = ns0; sw1 = ns1;
    }
}

// ---------------- host: constant precomputation ----------------
static inline void m2mul(const double* X, const double* Y, double* Z) {
    double z0 = X[0]*Y[0] + X[1]*Y[2];
    double z1 = X[0]*Y[1] + X[1]*Y[3];
    double z2 = X[2]*Y[0] + X[3]*Y[2];
    double z3 = X[2]*Y[1] + X[3]*Y[3];
    Z[0]=z0; Z[1]=z1; Z[2]=z2; Z[3]=z3;
}
static inline void m2vec(const double* X, const double* v, double* o) {
    double o0 = X[0]*v[0] + X[1]*v[1];
    double o1 = X[2]*v[0] + X[3]*v[1];
    o[0]=o0; o[1]=o1;
}

extern "C" void kernel_launch(void* const* d_in, const int* in_sizes, int n_in,
                              void* d_out, int out_size, void* d_ws, size_t ws_size,
                              hipStream_t stream) {
    (void)in_sizes; (void)n_in; (void)out_size; (void)ws_size;
    const float* x = (const float*)d_in[0];
    float* out   = (float*)d_out;
    float* cblk  = (float*)d_ws;                 // 2 * NBLOCKS floats   (8 KB)
    float* sblk  = cblk + 2 * NBLOCKS;           // 2 * NBLOCKS floats   (8 KB)
    float* ewave = sblk + 2 * NBLOCKS;           // 16 * NBLOCKS floats  (64 KB)

    // filter constants (compile-time parameters, computed in double)
    const double pi = 3.14159265358979323846;
    double g    = tan(pi / 44100.0 * 1000.0);
    double twoR = 1.0 / 0.707;
    double T    = 1.0 / (1.0 + g * (g + twoR));
    double H[4]   = { T * 1.0, T * (-g), T * g, T * (twoR * g + 1.0) };
    double gHB[2] = { g * T, g * T * g };
    double A[4]   = { 2.0*H[0] - 1.0, 2.0*H[1], 2.0*H[2], 2.0*H[3] - 1.0 };
    double b[2]   = { 2.0*gHB[0], 2.0*gHB[1] };
    double h[2]   = { H[2], H[3] };          // output row (lowpass: Y[...,1])
    double d      = gHB[1];

    // A^(2^k), k = 0..13
    double ap[14][4];
    for (int i = 0; i < 4; ++i) ap[0][i] = A[i];
    for (int k = 1; k < 14; ++k) m2mul(ap[k-1], ap[k-1], ap[k]);

    FConsts C;
    for (int i = 0; i < 4; ++i) C.A[i] = (float)A[i];
    C.b2[0] = (float)b[0]; C.b2[1] = (float)b[1];
    for (int l = 0; l < 8; ++l) for (int i = 0; i < 4; ++i) C.P[l][i] = (float)ap[5+l][i];
    for (int l = 0; l < 4; ++l) for (int i = 0; i < 4; ++i) C.A16o[l][i] = (float)ap[4+l][i];
    for (int i = 0; i < 4; ++i) C.A256[i]  = (float)ap[8][i];
    for (int i = 0; i < 4; ++i) C.A8192[i] = (float)ap[13][i];
    {
        double M[4] = {1,0,0,1};
        for (int wv = 0; wv < 8; ++wv) {
            for (int i = 0; i < 4; ++i) C.A1024p[wv][i] = (float)M[i];
            m2mul(ap[10], M, M);
        }
    }
    {   // W[:,k] = A^(15-k) b
        double u[2] = { b[0], b[1] };
        for (int k = 15; k >= 0; --k) {
            C.W0[k] = (float)u[0]; C.W1[k] = (float)u[1];
            m2vec(A, u, u);
        }
    }
    {   // L[i][k]: h . (A^(i-1-k) b) for k<i; d on diag; 0 above
        double ue[15][2];
        ue[0][0] = b[0]; ue[0][1] = b[1];
        for (int e = 1; e < 15; ++e) m2vec(A, ue[e-1], ue[e]);
        for (int i = 0; i < 16; ++i)
            for (int k = 0; k < 16; ++k) {
                if (k < i)       C.L[i][k] = (float)(h[0]*ue[i-1-k][0] + h[1]*ue[i-1-k][1]);
                else if (k == i) C.L[i][k] = (float)d;
                else             C.L[i][k] = 0.f;
            }
    }
    {   // G[i] = h^T A^i
        double M[4] = {1,0,0,1};
        for (int i = 0; i < 16; ++i) {
            C.G0[i] = (float)(h[0]*M[0] + h[1]*M[2]);
            C.G1[i] = (float)(h[0]*M[1] + h[1]*M[3]);
            m2mul(A, M, M);
        }
    }

    k_blockconst<<<dim3(NBLOCKS), dim3(256), 0, stream>>>(x, cblk, ewave, C);
    k_scanblocks<<<dim3(1), dim3(64), 0, stream>>>(cblk, sblk, C);
    k_output<<<dim3(NBLOCKS), dim3(256), 0, stream>>>(x, sblk, ewave, out, C);
}